// LSTMDecoder_13322988552521
// MI455X (gfx1250) — compile-verified
//
#include <hip/hip_runtime.h>
#include <hip/hip_bf16.h>
#include <math.h>

// ---------------------------------------------------------------------------
// LSTM decoder with attention (Show-Attend-Tell), fp32, CDNA5 WMMA version.
// All GEMMs use V_WMMA_F32_16X16X4_F32 (native fp32 matrix op on gfx1250).
// Each wave computes a 16x32 output strip (2 accumulators, shared A operand)
// so consecutive WMMAs are independent and the XDL pipe stays fed.
// ---------------------------------------------------------------------------

typedef float v2f __attribute__((ext_vector_type(2)));
typedef float v8f __attribute__((ext_vector_type(8)));

#define kB   32
#define kP   49
#define kENC 2048
#define kV   10000
#define kE   512
#define kH   512
#define kA   512
#define kT   60
#define kG4H 2048          // 4*H
#define kHC  2560          // H + ENC

#define KCH  32            // K-chunk staged in LDS per iteration
#define LDSS 36            // 32 + 4 pad: rows 16B-aligned (36*4=144), banks spread

static __device__ __forceinline__ float sigmoidf_(float x) {
    return 1.0f / (1.0f + __expf(-x));
}

// ---------------------------------------------------------------------------
// Generic GEMM: out = X[M,K] @ W[N,K]^T (+bias)(+bias2)(+addC[M,N])
// Block: 128 threads = 4 wave32s. Block tile 32(M) x 64(N).
// Wave w: M-offset (w&1)*16, N-offset (w>>1)*32, two 16x16 accumulators.
// outMode==1 remaps row=(t*B+b) -> output offset (b*T+t)*N ([B,T,V] layout).
// ---------------------------------------------------------------------------
__global__ void gemm_xwt_kernel(const float* __restrict__ X,
                                const float* __restrict__ W,
                                const float* __restrict__ bias,
                                const float* __restrict__ bias2,
                                const float* __restrict__ addC,
                                float* __restrict__ out,
                                int M, int N, int K,
                                int outMode, int Td, int Bd) {
    __shared__ float sX[32 * LDSS];
    __shared__ float sW[64 * LDSS];

    const int tid  = threadIdx.x;
    const int lane = tid & 31;
    const int wv   = tid >> 5;            // wave id 0..3
    const int blockM = blockIdx.x * 32;
    const int blockN = blockIdx.y * 64;
    const int tm  = (wv & 1) * 16;        // wave's M offset within block tile
    const int tnw = (wv >> 1) * 32;       // wave's N offset within block tile
    const int l15  = lane & 15;
    const int koff = (lane >> 4) << 1;    // 0 for lanes 0-15, 2 for lanes 16-31

    v8f acc0 = {};
    v8f acc1 = {};

    for (int k0 = 0; k0 < K; k0 += KCH) {
        // ---- Cooperative staging, 128-bit transactions -------------------
        // X tile: 32 rows x 32 floats = 256 float4 (2 per thread)
        for (int i = tid; i < 256; i += 128) {
            int r  = i >> 3;
            int c4 = (i & 7) << 2;
            int gm = blockM + r;
            float4 v = make_float4(0.f, 0.f, 0.f, 0.f);
            if (gm < M) v = *(const float4*)&X[(size_t)gm * K + k0 + c4];
            *(float4*)&sX[r * LDSS + c4] = v;
        }
        // W tile: 64 rows x 32 floats = 512 float4 (4 per thread)
        for (int i = tid; i < 512; i += 128) {
            int r  = i >> 3;
            int c4 = (i & 7) << 2;
            int gn = blockN + r;
            float4 v = make_float4(0.f, 0.f, 0.f, 0.f);
            if (gn < N) v = *(const float4*)&W[(size_t)gn * K + k0 + c4];
            *(float4*)&sW[r * LDSS + c4] = v;
        }
        // ---- Prefetch next K-chunk while this one computes ---------------
        int kn = k0 + KCH;
        if (kn < K) {
            if (tid < 64) {
                int gn = blockN + tid;
                if (gn < N) __builtin_prefetch(&W[(size_t)gn * K + kn], 0, 0);
            } else if (tid < 96) {
                int gm = blockM + (tid - 64);
                if (gm < M) __builtin_prefetch(&X[(size_t)gm * K + kn], 0, 0);
            }
        }
        __syncthreads();

        const float* px  = &sX[(tm + l15) * LDSS];
        const float* pw0 = &sW[(tnw + l15) * LDSS];
        const float* pw1 = &sW[(tnw + 16 + l15) * LDSS];
        #pragma unroll
        for (int kk = 0; kk < KCH; kk += 4) {
            v2f a, b0, b1;
            a.x  = px[kk + koff];
            a.y  = px[kk + koff + 1];
            b0.x = pw0[kk + koff];
            b0.y = pw0[kk + koff + 1];
            b1.x = pw1[kk + koff];
            b1.y = pw1[kk + koff + 1];
            // Two independent WMMAs share the A operand.
            acc0 = __builtin_amdgcn_wmma_f32_16x16x4_f32(
                false, a, false, b0, (short)0, acc0, false, false);
            acc1 = __builtin_amdgcn_wmma_f32_16x16x4_f32(
                false, a, false, b1, (short)0, acc1, false, false);
        }
        __syncthreads();
    }

    // ---- Epilogue. C/D layout: VGPR r -> row (r | +8 lanes 16-31), col = lane&15.
    const int rbase = blockM + tm + ((lane >> 4) << 3);
    #pragma unroll
    for (int half = 0; half < 2; ++half) {
        const v8f& acc = half ? acc1 : acc0;
        int col = blockN + tnw + half * 16 + l15;
        if (col < N) {
            float bsum = 0.0f;
            if (bias)  bsum += bias[col];
            if (bias2) bsum += bias2[col];
            #pragma unroll
            for (int r = 0; r < 8; ++r) {
                int row = rbase + r;
                if (row < M) {
                    float v = acc[r] + bsum;
                    if (addC) v += addC[(size_t)row * N + col];
                    size_t o;
                    if (outMode == 0) {
                        o = (size_t)row * N + col;
                    } else {
                        int tt = row / Bd;
                        int bb = row - tt * Bd;
                        o = ((size_t)(bb * Td + tt)) * N + col;
                    }
                    out[o] = v;
                }
            }
        }
    }
}

// ---------------------------------------------------------------------------
// Gather embeddings into GEMM layout: emb_mat[(t*B+b), e] = table[captions[b,t], e]
// ---------------------------------------------------------------------------
__global__ void embed_gather_kernel(const int* __restrict__ captions,
                                    const float* __restrict__ table,
                                    float* __restrict__ emb_mat) {
    int idx = blockIdx.x * blockDim.x + threadIdx.x;
    if (idx >= kT * kB * kE) return;
    int e   = idx & (kE - 1);
    int row = idx >> 9;            // t*B + b
    int t   = row >> 5;
    int b   = row & 31;
    int tok = captions[b * kT + t];
    emb_mat[idx] = table[(size_t)tok * kE + e];
}

__global__ void combine_bias_kernel(const float* __restrict__ b_ih,
                                    const float* __restrict__ b_hh,
                                    float* __restrict__ comb) {
    int n = blockIdx.x * blockDim.x + threadIdx.x;
    if (n < kG4H) comb[n] = b_ih[n] + b_hh[n];
}

// ---------------------------------------------------------------------------
// LSTM cell elementwise. gates[b, 4H] split into i,f,g,o chunks of H.
// Writes h into hbuf (recurrence) and into hc_buf[t*B+b, 0:H] (deferred W_out).
// ---------------------------------------------------------------------------
__global__ void lstm_cell_kernel(const float* __restrict__ gates,
                                 float* __restrict__ cbuf,
                                 float* __restrict__ hbuf,
                                 float* __restrict__ hc_buf,
                                 int t) {
    int idx = blockIdx.x * blockDim.x + threadIdx.x;
    if (idx >= kB * kH) return;
    int b = idx >> 9;
    int j = idx & (kH - 1);
    const float* g = gates + (size_t)b * kG4H;
    float gi = g[j];
    float gf = g[kH + j];
    float gg = g[2 * kH + j];
    float go = g[3 * kH + j];
    float cn = sigmoidf_(gf) * cbuf[idx] + sigmoidf_(gi) * tanhf(gg);
    cbuf[idx] = cn;
    float hn = sigmoidf_(go) * tanhf(cn);
    hbuf[idx] = hn;
    hc_buf[((size_t)(t * kB + b)) * kHC + j] = hn;
}

// ---------------------------------------------------------------------------
// Attention: scores -> softmax over P=49 -> context. One block per batch b.
// Writes context into hc_buf[t*B+b, H:H+ENC].
// ---------------------------------------------------------------------------
__global__ void attention_kernel(const float* __restrict__ features,
                                 const float* __restrict__ f_i,
                                 const float* __restrict__ att_h,
                                 const float* __restrict__ Va,
                                 const float* __restrict__ ba_v,
                                 float* __restrict__ hc_buf,
                                 int t) {
    __shared__ float s_score[kP];
    __shared__ float s_alpha[kP];
    const int b   = blockIdx.x;
    const int tid = threadIdx.x;

    // scores[p] = Va . tanh(att_h[b] + f_i[b,p,:]) + ba_v
    for (int p = tid; p < kP; p += blockDim.x) {
        const float* fi = f_i + ((size_t)(b * kP + p)) * kA;
        const float* ah = att_h + (size_t)b * kA;
        float s = ba_v[0];
        for (int a = 0; a < kA; ++a) {
            s += Va[a] * tanhf(ah[a] + fi[a]);
        }
        s_score[p] = s;
    }
    __syncthreads();

    if (tid == 0) {
        float m = s_score[0];
        for (int p = 1; p < kP; ++p) m = fmaxf(m, s_score[p]);
        float sum = 0.0f;
        for (int p = 0; p < kP; ++p) {
            float e = __expf(s_score[p] - m);
            s_alpha[p] = e;
            sum += e;
        }
        float inv = 1.0f / sum;
        for (int p = 0; p < kP; ++p) s_alpha[p] *= inv;
    }
    __syncthreads();

    // context[e] = sum_p alpha[p] * features[b,p,e]
    float* ctx = hc_buf + ((size_t)(t * kB + b)) * kHC + kH;
    for (int e = tid; e < kENC; e += blockDim.x) {
        const float* fb = features + ((size_t)b * kP) * kENC + e;
        float acc = 0.0f;
        for (int p = 0; p < kP; ++p) {
            acc += s_alpha[p] * fb[(size_t)p * kENC];
        }
        ctx[e] = acc;
    }
}

// ---------------------------------------------------------------------------
// Host launcher
// ---------------------------------------------------------------------------
extern "C" void kernel_launch(void* const* d_in, const int* in_sizes, int n_in,
                              void* d_out, int out_size, void* d_ws, size_t ws_size,
                              hipStream_t stream) {
    const float* features = (const float*)d_in[0];   // [B,P,ENC]
    const float* pooled   = (const float*)d_in[1];   // [B,ENC]
    const int*   captions = (const int*)d_in[2];     // [B,T]
    const float* table    = (const float*)d_in[3];   // [V,E]
    const float* W_ih     = (const float*)d_in[4];   // [4H,E]
    const float* b_ih     = (const float*)d_in[5];
    const float* W_hh     = (const float*)d_in[6];   // [4H,H]
    const float* b_hh     = (const float*)d_in[7];
    const float* Wa_h     = (const float*)d_in[8];   // [A,H]
    const float* ba_h     = (const float*)d_in[9];
    const float* Wa_f     = (const float*)d_in[10];  // [A,ENC]
    const float* ba_f     = (const float*)d_in[11];
    const float* Va       = (const float*)d_in[12];  // [1,A]
    const float* ba_v     = (const float*)d_in[13];  // [1]
    const float* W_init_h = (const float*)d_in[14];  // [H,ENC]
    const float* b_init_h = (const float*)d_in[15];
    const float* W_init_c = (const float*)d_in[16];  // [H,ENC]
    const float* b_init_c = (const float*)d_in[17];
    const float* W_out    = (const float*)d_in[18];  // [V,H+ENC]
    const float* b_out    = (const float*)d_in[19];
    float* out = (float*)d_out;                      // [B,T,V]

    // Workspace carve-up (floats)
    float* ws      = (float*)d_ws;
    float* emb_mat = ws;                                   // [T*B, E]
    float* x_proj  = emb_mat + (size_t)kT * kB * kE;       // [T*B, 4H]
    float* f_i     = x_proj + (size_t)kT * kB * kG4H;      // [B*P, A]
    float* gates   = f_i + (size_t)kB * kP * kA;           // [B, 4H]
    float* hbuf    = gates + (size_t)kB * kG4H;            // [B, H]
    float* cbuf    = hbuf + (size_t)kB * kH;               // [B, H]
    float* att_h   = cbuf + (size_t)kB * kH;               // [B, A]
    float* hc_buf  = att_h + (size_t)kB * kA;              // [T*B, H+ENC]
    float* comb    = hc_buf + (size_t)kT * kB * kHC;       // [4H]

    const dim3 blk128(128);
    const dim3 blk256(256);

    // ---- Phase 0: hoisted precomputes ------------------------------------
    embed_gather_kernel<<<dim3((kT * kB * kE + 255) / 256), blk256, 0, stream>>>(
        captions, table, emb_mat);
    combine_bias_kernel<<<dim3((kG4H + 255) / 256), blk256, 0, stream>>>(
        b_ih, b_hh, comb);

    // x_proj[(t*B+b), :] = emb @ W_ih^T + (b_ih + b_hh)   [1920,512]x[512,2048]
    gemm_xwt_kernel<<<dim3((kT * kB) / 32, kG4H / 64), blk128, 0, stream>>>(
        emb_mat, W_ih, comb, nullptr, nullptr, x_proj,
        kT * kB, kG4H, kE, 0, 0, 0);

    // h0 / c0 : [32,2048]x[2048,512]
    gemm_xwt_kernel<<<dim3(1, kH / 64), blk128, 0, stream>>>(
        pooled, W_init_h, b_init_h, nullptr, nullptr, hbuf,
        kB, kH, kENC, 0, 0, 0);
    gemm_xwt_kernel<<<dim3(1, kH / 64), blk128, 0, stream>>>(
        pooled, W_init_c, b_init_c, nullptr, nullptr, cbuf,
        kB, kH, kENC, 0, 0, 0);

    // f_i = features @ Wa_f^T + ba_f : [1568,2048]x[2048,512]
    gemm_xwt_kernel<<<dim3((kB * kP) / 32, kA / 64), blk128, 0, stream>>>(
        features, Wa_f, ba_f, nullptr, nullptr, f_i,
        kB * kP, kA, kENC, 0, 0, 0);

    // ---- Phase 1: sequential recurrence ----------------------------------
    for (int t = 0; t < kT; ++t) {
        // gates = h @ W_hh^T + x_proj[t]   [32,512]x[512,2048]
        gemm_xwt_kernel<<<dim3(1, kG4H / 64), blk128, 0, stream>>>(
            hbuf, W_hh, nullptr, nullptr, x_proj + (size_t)t * kB * kG4H, gates,
            kB, kG4H, kH, 0, 0, 0);

        lstm_cell_kernel<<<dim3((kB * kH + 255) / 256), blk256, 0, stream>>>(
            gates, cbuf, hbuf, hc_buf, t);

        // att_h = h @ Wa_h^T + ba_h   [32,512]x[512,512]
        gemm_xwt_kernel<<<dim3(1, kA / 64), blk128, 0, stream>>>(
            hbuf, Wa_h, ba_h, nullptr, nullptr, att_h,
            kB, kA, kH, 0, 0, 0);

        attention_kernel<<<dim3(kB), blk256, 0, stream>>>(
            features, f_i, att_h, Va, ba_v, hc_buf, t);
    }

    // ---- Phase 2: one big deferred output projection ---------------------
    // logits = hc_buf @ W_out^T + b_out : [1920,2560]x[2560,10000],
    // stored directly as [B,T,V] via outMode=1.
    gemm_xwt_kernel<<<dim3((kT * kB) / 32, (kV + 63) / 64), blk128, 0, stream>>>(
        hc_buf, W_out, b_out, nullptr, nullptr, out,
        kT * kB, kV, kHC, 1, kT, kB);
}